// GAT_53197464928924
// MI455X (gfx1250) — compile-verified
//
#include <hip/hip_runtime.h>
#include <hip/hip_bf16.h>
#include <cmath>

#define NNODES   10000
#define IN_CH    128
#define HID      64
#define OUT_CH   40
#define HEADS    8
#define NEG_SLOPE 0.2f

typedef float v2f __attribute__((ext_vector_type(2)));
typedef float v8f __attribute__((ext_vector_type(8)));

// ---------------------------------------------------------------------------
// atomics
// ---------------------------------------------------------------------------
__device__ __forceinline__ void atomic_max_f32(float* addr, float val) {
  if (val >= 0.0f) atomicMax((int*)addr, __float_as_int(val));
  else             atomicMin((unsigned int*)addr, __float_as_uint(val));
}

__device__ __forceinline__ void atomic_add_f32(float* p, float v) {
  unsafeAtomicAdd(p, v);   // native global_atomic_add_f32
}

// ---------------------------------------------------------------------------
// WMMA f32 GEMM: C[M,N] = act(A[M,K] @ B[K,N] + bias)
// One wave computes a 16 x (16*NT) strip with NT v8f accumulators.
// * A/B addressed as base + unsigned 32-bit offset -> global_load saddr form
// * software-pipelined: fragments for k+4 are fetched while k's WMMAs issue
// * GUARD: clamp B column address; OOB columns only pollute D columns that
//   are never stored, so no value masking is needed.
// ---------------------------------------------------------------------------
template<int NT, bool GUARD, int ACT>
__global__ void __launch_bounds__(32)
wmma_gemm_f32(const float* __restrict__ A, const float* __restrict__ B,
              const float* __restrict__ bias, float* __restrict__ C,
              int M, int N, int K)
{
  const int lane = threadIdx.x;            // 0..31, full wave
  const int m0   = blockIdx.x * 16;
  const int n0   = blockIdx.y * (16 * NT);
  const int r    = lane & 15;              // A row / B-D column within tile
  const int koff = (lane >> 4) << 1;       // lanes 0-15 -> K {0,1}; 16-31 -> {2,3}
  const unsigned uN = (unsigned)N;

  int      col[NT];
  unsigned cc[NT];                         // clamped column (load-safe)
#pragma unroll
  for (int t = 0; t < NT; ++t) {
    col[t] = n0 + 16 * t + r;
    cc[t]  = GUARD ? (unsigned)(col[t] < N ? col[t] : N - 1) : (unsigned)col[t];
  }

  unsigned aoff = (unsigned)(m0 + r) * (unsigned)K + (unsigned)koff;
  unsigned boff = (unsigned)koff * uN;

  v8f acc[NT];
#pragma unroll
  for (int t = 0; t < NT; ++t) acc[t] = v8f{0.f,0.f,0.f,0.f,0.f,0.f,0.f,0.f};

  // ---- prologue: fetch k=0 fragments ----
  v2f a0, b0[NT];
  a0.x = A[aoff];
  a0.y = A[aoff + 1];
#pragma unroll
  for (int t = 0; t < NT; ++t) {
    b0[t].x = B[boff + cc[t]];
    b0[t].y = B[boff + uN + cc[t]];
  }

  for (int k = 0; k < K - 4; k += 4) {
    // fetch next iteration's fragments while current WMMAs execute
    const unsigned an = aoff + 4;
    const unsigned bn = boff + 4u * uN;
    v2f a1, b1[NT];
    a1.x = A[an];
    a1.y = A[an + 1];
#pragma unroll
    for (int t = 0; t < NT; ++t) {
      b1[t].x = B[bn + cc[t]];
      b1[t].y = B[bn + uN + cc[t]];
    }
#pragma unroll
    for (int t = 0; t < NT; ++t)
      acc[t] = __builtin_amdgcn_wmma_f32_16x16x4_f32(
          false, a0, false, b0[t], (short)0, acc[t], false, false);
    a0 = a1;
#pragma unroll
    for (int t = 0; t < NT; ++t) b0[t] = b1[t];
    aoff = an;
    boff = bn;
  }

  // ---- epilogue: last K-step ----
#pragma unroll
  for (int t = 0; t < NT; ++t)
    acc[t] = __builtin_amdgcn_wmma_f32_16x16x4_f32(
        false, a0, false, b0[t], (short)0, acc[t], false, false);

  const int half = lane >> 4;              // D: M = vgpr + 8*half
#pragma unroll
  for (int t = 0; t < NT; ++t) {
    if (GUARD && col[t] >= N) continue;    // divergent skip after all WMMAs
    const float bv = bias ? bias[cc[t]] : 0.f;
#pragma unroll
    for (int i = 0; i < 8; ++i) {
      float v = acc[t][i] + bv;
      if (ACT == 1) v = v > 0.f ? v : 0.f;
      C[(unsigned)(m0 + i + half * 8) * uN + (unsigned)col[t]] = v;
    }
  }
}

// ---------------------------------------------------------------------------
// attention dot products per (node, head)
// ---------------------------------------------------------------------------
__global__ void node_alpha(const float* __restrict__ h,
                           const float* __restrict__ att_s,
                           const float* __restrict__ att_d,
                           float* __restrict__ asrc, float* __restrict__ adst,
                           int Nn, int H, int C)
{
  unsigned idx = blockIdx.x * blockDim.x + threadIdx.x;
  if (idx >= (unsigned)(Nn * H)) return;
  const int n = idx / H, hh = idx % H;
  const float* hp = h + (size_t)n * H * C + (size_t)hh * C;
  const float* as = att_s + (size_t)hh * C;
  const float* ad = att_d + (size_t)hh * C;
  float ss = 0.f, sd = 0.f;
  for (int c = 0; c < C; ++c) { const float v = hp[c]; ss += v * as[c]; sd += v * ad[c]; }
  asrc[idx] = ss; adst[idx] = sd;
}

// ---------------------------------------------------------------------------
// pass A: edge score = leaky_relu(asrc[src]+adst[dst]); segment max by dst
// ---------------------------------------------------------------------------
__global__ void edge_score_max(const int* __restrict__ ei,
                               const float* __restrict__ asrc,
                               const float* __restrict__ adst,
                               float* __restrict__ score,
                               float* __restrict__ smax,
                               int E, int Etot, int H)
{
  unsigned gid = blockIdx.x * blockDim.x + threadIdx.x;
  if (gid >= (unsigned)Etot * (unsigned)H) return;
  const int e  = gid / H;
  const int hh = gid % H;
  int s, d;
  if (e < E) { s = ei[e]; d = ei[E + e]; } else { s = d = e - E; }
  float sc = asrc[(size_t)s * H + hh] + adst[(size_t)d * H + hh];
  sc = sc > 0.f ? sc : NEG_SLOPE * sc;
  score[gid] = sc;
  atomic_max_f32(&smax[(size_t)d * H + hh], sc);
}

__global__ void smax_fixup(float* __restrict__ p, int n)
{
  unsigned i = blockIdx.x * blockDim.x + threadIdx.x;
  if (i >= (unsigned)n) return;
  const float v = p[i];
  if (!isfinite(v)) p[i] = 0.f;
}

// ---------------------------------------------------------------------------
// pass B: ex = exp(score - smax[dst]); segment sum of ex by dst; score <- ex
// ---------------------------------------------------------------------------
__global__ void edge_exp_sum(const int* __restrict__ ei,
                             float* __restrict__ score,
                             const float* __restrict__ smax,
                             float* __restrict__ denom,
                             int E, int Etot, int H)
{
  unsigned gid = blockIdx.x * blockDim.x + threadIdx.x;
  if (gid >= (unsigned)Etot * (unsigned)H) return;
  const int e  = gid / H;
  const int hh = gid % H;
  int d;
  if (e < E) { d = ei[E + e]; } else { d = e - E; }
  const float ex = __expf(score[gid] - smax[(size_t)d * H + hh]);
  score[gid] = ex;
  atomic_add_f32(&denom[(size_t)d * H + hh], ex);
}

// ---------------------------------------------------------------------------
// out rows pre-seeded with bias (so atomics land on top of it)
// ---------------------------------------------------------------------------
__global__ void init_rows(float* __restrict__ out, const float* __restrict__ bias,
                          int rows, int L)
{
  unsigned gid = blockIdx.x * blockDim.x + threadIdx.x;
  if (gid >= (unsigned)rows * (unsigned)L) return;
  out[gid] = bias[gid % L];
}

__global__ void fill_f32(float* __restrict__ p, float v, int n)
{
  unsigned i = blockIdx.x * blockDim.x + threadIdx.x;
  if (i < (unsigned)n) p[i] = v;
}

// ---------------------------------------------------------------------------
// pass C: out[dst] += h[src] * (ex / (denom[dst]+eps))
// one thread per (edge, 4 consecutive channels): b128 gather, 4 f32 atomics
// ---------------------------------------------------------------------------
__global__ void edge_aggregate_v4(const int* __restrict__ ei,
                                  const float* __restrict__ hc,
                                  const float* __restrict__ ex,
                                  const float* __restrict__ denom,
                                  float* __restrict__ out,
                                  int E, int Etot, int H, int C)
{
  const unsigned HC  = (unsigned)(H * C);
  const unsigned HC4 = HC >> 2;
  unsigned gid = blockIdx.x * blockDim.x + threadIdx.x;
  if (gid >= (unsigned)Etot * HC4) return;
  const int e  = gid / HC4;
  const int q  = gid - (unsigned)e * HC4;      // float4 index within row
  const int hh = (q << 2) / C;                 // 4 consecutive channels share a head
  int s, d;
  if (e < E) { s = ei[e]; d = ei[E + e]; } else { s = d = e - E; }
  const float alpha = ex[(size_t)e * H + hh] / (denom[(size_t)d * H + hh] + 1e-16f);
  const float4 v = ((const float4*)(hc + (size_t)s * HC))[q];
  float* op = out + (size_t)d * HC + ((size_t)q << 2);
  atomic_add_f32(op + 0, v.x * alpha);
  atomic_add_f32(op + 1, v.y * alpha);
  atomic_add_f32(op + 2, v.z * alpha);
  atomic_add_f32(op + 3, v.w * alpha);
}

__global__ void elu_inplace(float* __restrict__ p, int n)
{
  unsigned i = blockIdx.x * blockDim.x + threadIdx.x;
  if (i >= (unsigned)n) return;
  const float v = p[i];
  p[i] = v > 0.f ? v : expm1f(v);
}

// ---------------------------------------------------------------------------
extern "C" void kernel_launch(void* const* d_in, const int* in_sizes, int n_in,
                              void* d_out, int out_size, void* d_ws, size_t ws_size,
                              hipStream_t stream)
{
  const float* x        = (const float*)d_in[0];
  const int*   ei       = (const int*)  d_in[1];
  const float* W1       = (const float*)d_in[2];
  const float* b1       = (const float*)d_in[3];
  const float* Wc1      = (const float*)d_in[4];
  const float* att_src1 = (const float*)d_in[5];
  const float* att_dst1 = (const float*)d_in[6];
  const float* bc1      = (const float*)d_in[7];
  const float* Wc2      = (const float*)d_in[8];
  const float* att_src2 = (const float*)d_in[9];
  const float* att_dst2 = (const float*)d_in[10];
  const float* bc2      = (const float*)d_in[11];
  const float* W2       = (const float*)d_in[12];
  const float* b2       = (const float*)d_in[13];
  float*       out      = (float*)d_out;

  const int E    = in_sizes[1] / 2;     // 320000
  const int Etot = E + NNODES;          // + self loops
  const int Nn   = NNODES;

  // ---- workspace layout (floats) ----
  float* ws   = (float*)d_ws;
  float* h0   = ws;                         // Nn*HID            (reused as hc2)
  float* hc1  = h0   + (size_t)Nn * HID;    // Nn*HEADS*HID
  float* out1 = hc1  + (size_t)Nn * HEADS * HID;   // Nn*HEADS*HID
  float* sc1  = out1 + (size_t)Nn * HEADS * HID;   // Etot*HEADS  (reused as sc2)
  float* as1  = sc1  + (size_t)Etot * HEADS;       // Nn*HEADS
  float* ad1  = as1  + (size_t)Nn * HEADS;
  float* mx1  = ad1  + (size_t)Nn * HEADS;
  float* dn1  = mx1  + (size_t)Nn * HEADS;
  float* as2  = dn1  + (size_t)Nn * HEADS;         // Nn
  float* ad2  = as2  + (size_t)Nn;
  float* mx2  = ad2  + (size_t)Nn;
  float* dn2  = mx2  + (size_t)Nn;
  float* out2 = dn2  + (size_t)Nn;                 // Nn*HID

  const int BT = 256;
  auto g1 = [&](long long n) { return dim3((unsigned)((n + BT - 1) / BT)); };

  // ===== lin0: h0 = relu(x @ W1 + b1)  [10000,128]x[128,64] =====
  wmma_gemm_f32<4, false, 1><<<dim3(Nn / 16, HID / 64), dim3(32), 0, stream>>>(
      x, W1, b1, h0, Nn, HID, IN_CH);

  // ===== conv1 projection: hc1 = h0 @ Wc1  [10000,64]x[64,512] =====
  wmma_gemm_f32<4, false, 0><<<dim3(Nn / 16, (HEADS * HID) / 64), dim3(32), 0, stream>>>(
      h0, Wc1, nullptr, hc1, Nn, HEADS * HID, HID);

  // attention dots
  node_alpha<<<g1((long long)Nn * HEADS), BT, 0, stream>>>(
      hc1, att_src1, att_dst1, as1, ad1, Nn, HEADS, HID);

  // segment softmax over edges (by dst)
  fill_f32<<<g1((long long)Nn * HEADS), BT, 0, stream>>>(mx1, -INFINITY, Nn * HEADS);
  fill_f32<<<g1((long long)Nn * HEADS), BT, 0, stream>>>(dn1, 0.f, Nn * HEADS);
  edge_score_max<<<g1((long long)Etot * HEADS), BT, 0, stream>>>(
      ei, as1, ad1, sc1, mx1, E, Etot, HEADS);
  smax_fixup<<<g1((long long)Nn * HEADS), BT, 0, stream>>>(mx1, Nn * HEADS);
  edge_exp_sum<<<g1((long long)Etot * HEADS), BT, 0, stream>>>(
      ei, sc1, mx1, dn1, E, Etot, HEADS);

  // aggregate messages (+bc1 seeded), then ELU
  init_rows<<<g1((long long)Nn * HEADS * HID), BT, 0, stream>>>(
      out1, bc1, Nn, HEADS * HID);
  edge_aggregate_v4<<<g1((long long)Etot * (HEADS * HID / 4)), BT, 0, stream>>>(
      ei, hc1, sc1, dn1, out1, E, Etot, HEADS, HID);
  elu_inplace<<<g1((long long)Nn * HEADS * HID), BT, 0, stream>>>(
      out1, Nn * HEADS * HID);

  // ===== conv2 projection: hc2 = out1 @ Wc2  [10000,512]x[512,64] =====
  float* hc2 = h0;  // reuse
  wmma_gemm_f32<4, false, 0><<<dim3(Nn / 16, HID / 64), dim3(32), 0, stream>>>(
      out1, Wc2, nullptr, hc2, Nn, HID, HEADS * HID);

  node_alpha<<<g1((long long)Nn), BT, 0, stream>>>(
      hc2, att_src2, att_dst2, as2, ad2, Nn, 1, HID);

  float* sc2 = sc1;  // reuse
  fill_f32<<<g1((long long)Nn), BT, 0, stream>>>(mx2, -INFINITY, Nn);
  fill_f32<<<g1((long long)Nn), BT, 0, stream>>>(dn2, 0.f, Nn);
  edge_score_max<<<g1((long long)Etot), BT, 0, stream>>>(
      ei, as2, ad2, sc2, mx2, E, Etot, 1);
  smax_fixup<<<g1((long long)Nn), BT, 0, stream>>>(mx2, Nn);
  edge_exp_sum<<<g1((long long)Etot), BT, 0, stream>>>(
      ei, sc2, mx2, dn2, E, Etot, 1);

  init_rows<<<g1((long long)Nn * HID), BT, 0, stream>>>(out2, bc2, Nn, HID);
  edge_aggregate_v4<<<g1((long long)Etot * (HID / 4)), BT, 0, stream>>>(
      ei, hc2, sc2, dn2, out2, E, Etot, 1, HID);

  // ===== final: out = out2 @ W2 + b2  [10000,64]x[64,40] (guard N=40) =====
  wmma_gemm_f32<4, true, 0><<<dim3(Nn / 16, 1), dim3(32), 0, stream>>>(
      out2, W2, b2, out, Nn, OUT_CH, HID);
}